// PytorchTokenFeaturer_30743375905274
// MI455X (gfx1250) — compile-verified
//
#include <hip/hip_runtime.h>

// ---------------------------------------------------------------------------
// Ragged segment featurizer: out[s] = concat(first, mean, max) over values rows
// values: [T,128] f32, offsets: [S] i32 (sorted, contiguous segments)
// out: [S, 384] f32
//
// MI455X (gfx1250): HBM-bound (~637 MB @ 23.3 TB/s ~= 27us). One wave32 per
// segment, lane = float4 column slice, rows streamed via CDNA5 async
// global->LDS copies (ASYNCcnt-pipelined, depth 6, 8 slots => ~3KB in flight
// per wave for latency hiding at the HBM roofline).
// ---------------------------------------------------------------------------

typedef float v4f __attribute__((ext_vector_type(4)));

#define WAVES_PER_BLOCK 8
#define DEPTH 6   // async issue-ahead depth (must be < SLOTS)
#define SLOTS 8   // LDS row slots per wave (power of 2, > DEPTH for WAR safety)

// One wave moves one 512B row: each lane copies 16B to its LDS slot address.
__device__ __forceinline__ void async_row_b128(unsigned lds_addr, const float* g) {
    asm volatile("global_load_async_to_lds_b128 %0, %1, off"
                 :: "v"(lds_addr), "v"((unsigned long long)(size_t)g)
                 : "memory");
}

#define WAIT_ASYNC(N) asm volatile("s_wait_asynccnt %0" :: "n"(N) : "memory")

__global__ __launch_bounds__(256) void token_featurizer_kernel(
    const float* __restrict__ values,
    const int*   __restrict__ offsets,
    float*       __restrict__ out,
    int T, int S)
{
    __shared__ v4f smem[WAVES_PER_BLOCK * SLOTS * 32];   // 32 KB / block

    const unsigned tid  = threadIdx.x;
    const unsigned wave = tid >> 5;
    const unsigned lane = tid & 31u;
    const int s = (int)(blockIdx.x * WAVES_PER_BLOCK + wave);
    if (s >= S) return;                    // wave-uniform; active waves keep EXEC all-ones

    const int off0 = offsets[s];
    const int off1 = (s + 1 < S) ? offsets[s + 1] : T;
    const int len  = off1 - off0;          // >= 1 by construction

    v4f* slotBase = &smem[(wave * SLOTS) * 32 + lane];
    const float* rowBase = values + (size_t)off0 * 128 + lane * 4;

    // ---- prologue: prime the async pipeline -------------------------------
    const int pre = (len < DEPTH) ? len : DEPTH;
    for (int i = 0; i < pre; ++i) {
        unsigned lds = (unsigned)(size_t)(slotBase + (unsigned)(i & (SLOTS - 1)) * 32);
        async_row_b128(lds, rowBase + (size_t)i * 128);
    }

    // ---- steady state: issue row k+DEPTH, wait, consume row k from LDS ----
    v4f first = {}, sum = {}, mx = {};
    for (int k = 0; k < len; ++k) {
        if (k + DEPTH < len) {
            unsigned slot = (unsigned)((k + DEPTH) & (SLOTS - 1));
            unsigned lds  = (unsigned)(size_t)(slotBase + slot * 32);
            async_row_b128(lds, rowBase + (size_t)(k + DEPTH) * 128);
            // k+DEPTH+1 issued, in-order completion => <=DEPTH-1 pending
            // guarantees rows <= k+1 have landed in LDS.
            WAIT_ASYNC(DEPTH - 1);
        } else {
            WAIT_ASYNC(0);                 // tail drain: everything issued is done
        }
        v4f r = slotBase[(unsigned)(k & (SLOTS - 1)) * 32];  // ds_load_b128
        if (k == 0) {
            first = r; sum = r; mx = r;
        } else {
            sum += r;
            mx.x = fmaxf(mx.x, r.x);
            mx.y = fmaxf(mx.y, r.y);
            mx.z = fmaxf(mx.z, r.z);
            mx.w = fmaxf(mx.w, r.w);
        }
    }

    // ---- epilogue: out[s] = [first | mean | max], write-once -> NT stores --
    const float inv = 1.0f / (float)len;
    v4f mean = sum * inv;
    float* o = out + (size_t)s * 384 + lane * 4;
    __builtin_nontemporal_store(first, (v4f*)(o));
    __builtin_nontemporal_store(mean,  (v4f*)(o + 128));
    __builtin_nontemporal_store(mx,    (v4f*)(o + 256));
}

extern "C" void kernel_launch(void* const* d_in, const int* in_sizes, int n_in,
                              void* d_out, int out_size, void* d_ws, size_t ws_size,
                              hipStream_t stream)
{
    const float* values  = (const float*)d_in[0];   // [T,128] f32
    // d_in[1]: segment_ids [T] i32 -- not needed (offsets + sortedness suffice)
    const int*   offsets = (const int*)d_in[2];     // [S] i32
    const int T = in_sizes[1];
    const int S = in_sizes[2];
    float* out = (float*)d_out;                     // [S,384] f32

    dim3 block(32 * WAVES_PER_BLOCK);
    dim3 grid((unsigned)((S + WAVES_PER_BLOCK - 1) / WAVES_PER_BLOCK));
    token_featurizer_kernel<<<grid, block, 0, stream>>>(values, offsets, out, T, S);
}